// NeighbourAssignment_80161269612940
// MI455X (gfx1250) — compile-verified
//
#include <hip/hip_runtime.h>
#include <hip/hip_bf16.h>

typedef float v8f __attribute__((ext_vector_type(8)));
typedef float v2f __attribute__((ext_vector_type(2)));

// ---------------------------------------------------------------------------
// Kernel 1: st[n,0:8] = x[n]@Ws + bs ; st[n,8:16] = x[n]@Wt + bt
// One thread per (node, o) pair; 16 lanes share a node -> broadcast x loads.
// ---------------------------------------------------------------------------
__global__ __launch_bounds__(256) void feast_proj(
    const float* __restrict__ x,
    const float* __restrict__ Ws, const float* __restrict__ bs,
    const float* __restrict__ Wt, const float* __restrict__ bt,
    float* __restrict__ st, int n_nodes)
{
    int tid  = blockIdx.x * blockDim.x + threadIdx.x;
    int node = tid >> 4;
    int o    = tid & 15;
    if (node >= n_nodes) return;
    const float* xr = x + (size_t)node * 64;
    float acc;
    if (o < 8) {
        acc = bs[o];
        #pragma unroll 8
        for (int c = 0; c < 64; ++c) acc = fmaf(xr[c], Ws[c * 8 + o], acc);
    } else {
        int oo = o - 8;
        acc = bt[oo];
        #pragma unroll 8
        for (int c = 0; c < 64; ++c) acc = fmaf(xr[c], Wt[c * 8 + oo], acc);
    }
    st[(size_t)node * 16 + o] = acc;
}

// ---------------------------------------------------------------------------
// Kernel 2: per 16-node block:
//   phase A: accumulate Gext rows (512 G values + 8 Qsum) in VGPRs -> LDS
//   phase B: out[16,64] = Gext[16,520] @ Wext[520,64] * (1/8) via WMMA f32
// dst is sorted, so edge ranges per node come from binary search.
// LDS row padded to 544 floats (17 regs * 32 lanes).
// ---------------------------------------------------------------------------
__global__ __launch_bounds__(128) void feast_main(
    const float* __restrict__ x,
    const int*   __restrict__ src,
    const int*   __restrict__ dst,
    const float* __restrict__ W,   // [8,64,64] == Wext rows 0..511
    const float* __restrict__ b,   // [8,64]    == Wext rows 512..519
    const float* __restrict__ st,  // [N,16]
    float*       __restrict__ out, // [N,64]
    int n_nodes, int n_edges)
{
    __shared__ float G[16 * 544];
    __shared__ int   bounds[17];

    const int t    = threadIdx.x;
    const int wave = t >> 5;
    const int lane = t & 31;
    const int i0   = blockIdx.x * 16;

    // Edge-range lower bounds for the 16 dst nodes of this block.
    if (t < 17) {
        int key = i0 + t;
        int lo = 0, hi = n_edges;
        while (lo < hi) {
            int mid = (lo + hi) >> 1;
            if (dst[mid] < key) lo = mid + 1; else hi = mid;
        }
        bounds[t] = lo;
    }
    __syncthreads();

    // ---------------- Phase A: edge accumulation -------------------------
    // Wave w owns rows w, w+4, w+8, w+12. Lane L owns Gext elements L+32k.
    //  k=2j   -> head j, channel L      (v = L + 64j)
    //  k=2j+1 -> head j, channel L+32
    //  k=16   -> Qsum[L] for L<8 (v = 512+L); lanes 8..31 write pad (unused)
    for (int u = 0; u < 4; ++u) {
        const int r = wave + 4 * u;
        const int i = i0 + r;
        float acc[17];
        #pragma unroll
        for (int k = 0; k < 17; ++k) acc[k] = 0.0f;

        if (i < n_nodes) {
            const float s_lane = st[(size_t)i * 16 + (lane & 7)];
            const int elo = bounds[r];
            const int ehi = bounds[r + 1];
            for (int e = elo; e < ehi; ++e) {
                const int j = __builtin_amdgcn_readfirstlane(src[e]);
                // softmax over 8 heads, replicated in each 8-lane subgroup
                float logit = s_lane + st[(size_t)j * 16 + 8 + (lane & 7)];
                float mx = logit;
                mx = fmaxf(mx, __shfl_xor(mx, 1, 32));
                mx = fmaxf(mx, __shfl_xor(mx, 2, 32));
                mx = fmaxf(mx, __shfl_xor(mx, 4, 32));
                float ev = __expf(logit - mx);
                float sm = ev;
                sm += __shfl_xor(sm, 1, 32);
                sm += __shfl_xor(sm, 2, 32);
                sm += __shfl_xor(sm, 4, 32);
                const float qn = ev / sm;

                float q[8];
                #pragma unroll
                for (int m = 0; m < 8; ++m) q[m] = __shfl(qn, m, 32);

                const float xa = x[(size_t)j * 64 + lane];
                const float xb = x[(size_t)j * 64 + lane + 32];
                #pragma unroll
                for (int m = 0; m < 8; ++m) {
                    acc[2 * m]     = fmaf(q[m], xa, acc[2 * m]);
                    acc[2 * m + 1] = fmaf(q[m], xb, acc[2 * m + 1]);
                }
                if (lane < 8) acc[16] += qn;   // Qsum
            }
        }
        #pragma unroll
        for (int k = 0; k < 17; ++k)
            G[r * 544 + lane + 32 * k] = acc[k];
    }
    __syncthreads();

    // ---------------- Phase B: WMMA GEMM ---------------------------------
    // Wave w computes out[i0:i0+16, 16w:16w+16]. K = 520 -> 130 steps of 4.
    // A frag (16x4 f32): lanes 0-15 hold M=lane, K0/K1; lanes 16-31 K2/K3.
    // B frag (4x16 f32): lanes 0-15 hold N=lane, K0/K1; lanes 16-31 K2/K3.
    const int arow  = lane & 15;
    const int khalf = (lane >> 4) * 2;           // 0 or 2
    const int col   = 16 * wave + (lane & 15);   // output column 0..63

    v8f d = {0.f, 0.f, 0.f, 0.f, 0.f, 0.f, 0.f, 0.f};

    #pragma unroll 2
    for (int kk = 0; kk < 128; ++kk) {           // K rows 0..511 from W
        const int kb = kk * 4 + khalf;
        v2f a, bf;
        a.x  = G[arow * 544 + kb];
        a.y  = G[arow * 544 + kb + 1];
        bf.x = W[(size_t)kb * 64 + col];
        bf.y = W[(size_t)(kb + 1) * 64 + col];
        d = __builtin_amdgcn_wmma_f32_16x16x4_f32(
                false, a, false, bf, (short)0, d, false, false);
    }
    #pragma unroll
    for (int kk = 128; kk < 130; ++kk) {         // K rows 512..519 from bias
        const int kb = kk * 4 + khalf;
        v2f a, bf;
        a.x  = G[arow * 544 + kb];
        a.y  = G[arow * 544 + kb + 1];
        bf.x = b[(kb - 512) * 64 + col];
        bf.y = b[(kb - 511) * 64 + col];
        d = __builtin_amdgcn_wmma_f32_16x16x4_f32(
                false, a, false, bf, (short)0, d, false, false);
    }

    // D layout: VGPR g -> lanes 0-15: M=g, N=lane; lanes 16-31: M=8+g.
    #pragma unroll
    for (int g = 0; g < 8; ++g) {
        const int row = g + (lane >> 4) * 8;
        const int node = i0 + row;
        if (node < n_nodes)
            out[(size_t)node * 64 + col] = d[g] * 0.125f;
    }
}

// ---------------------------------------------------------------------------
extern "C" void kernel_launch(void* const* d_in, const int* in_sizes, int n_in,
                              void* d_out, int out_size, void* d_ws, size_t ws_size,
                              hipStream_t stream)
{
    const float* x   = (const float*)d_in[0];   // [N,64]
    const int*   src = (const int*)  d_in[1];   // [E]
    const int*   dst = (const int*)  d_in[2];   // [E] sorted
    const float* W   = (const float*)d_in[3];   // [8,64,64]
    const float* b   = (const float*)d_in[4];   // [8,64]
    const float* Ws  = (const float*)d_in[5];   // [64,8]
    const float* bs  = (const float*)d_in[6];   // [8]
    const float* Wt  = (const float*)d_in[7];   // [64,8]
    const float* bt  = (const float*)d_in[8];   // [8]
    float* out = (float*)d_out;

    const int n_nodes = in_sizes[0] / 64;
    const int n_edges = in_sizes[1];

    float* st = (float*)d_ws;                   // [N,16] scratch

    const int pthreads = 256;
    const int pgrid = (n_nodes * 16 + pthreads - 1) / pthreads;
    feast_proj<<<pgrid, pthreads, 0, stream>>>(x, Ws, bs, Wt, bt, st, n_nodes);

    const int blocks = (n_nodes + 15) / 16;
    feast_main<<<blocks, 128, 0, stream>>>(x, src, dst, W, b, st, out,
                                           n_nodes, n_edges);
}